// MultiHeadAttentionMIL_65103114272882
// MI455X (gfx1250) — compile-verified
//
#include <hip/hip_runtime.h>
#include <hip/hip_bf16.h>
#include <math.h>

typedef __attribute__((ext_vector_type(2))) float v2f;
typedef __attribute__((ext_vector_type(8))) float v8f;

#define N_TOT 131072
#define D 512
#define H 8
#define HD 64

// ---- gfx1250 async copy: 16B global -> LDS, tracked by ASYNCcnt ----
__device__ __forceinline__ void async_copy_b128(float* lds_dst, const float* gsrc) {
    unsigned           loff  = (unsigned)(size_t)lds_dst;           // LDS offset = flat addr[31:0]
    unsigned long long gaddr = (unsigned long long)(size_t)gsrc;
    asm volatile("global_load_async_to_lds_b128 %0, %1, off"
                 :: "v"(loff), "v"(gaddr) : "memory");
}
__device__ __forceinline__ void wait_async0() {
    asm volatile("s_wait_asynccnt 0x0" ::: "memory");
}

// ---------------- Kernel 0: prep  q, c_h, r_h ----------------
__global__ __launch_bounds__(512) void k_prep(
    const float* __restrict__ mq, const float* __restrict__ Wq, const float* __restrict__ bq,
    const float* __restrict__ Wk, const float* __restrict__ bk,
    float* __restrict__ c_out, float* __restrict__ r_out)
{
    __shared__ float mqs[D];
    __shared__ float qs[D];
    const int t = threadIdx.x;
    mqs[t] = mq[t];
    __syncthreads();
    // q[t] = bq[t] + Wq[t,:] . mq
    float acc = bq[t];
    const float* row = Wq + (size_t)t * D;
    for (int j = 0; j < D; ++j) acc = fmaf(row[j], mqs[j], acc);
    qs[t] = acc;
    __syncthreads();
    if (t < H) {
        float c = 0.f;
        for (int d2 = 0; d2 < HD; ++d2) c = fmaf(qs[t * HD + d2], bk[t * HD + d2], c);
        c_out[t] = c;
    }
    // r[h][j] = sum_d q[h*64+d] * Wk[h*64+d, j]   (coalesced over j = t)
    for (int h = 0; h < H; ++h) {
        float s = 0.f;
        for (int d2 = 0; d2 < HD; ++d2)
            s = fmaf(qs[h * HD + d2], Wk[(size_t)(h * HD + d2) * D + t], s);
        r_out[h * D + t] = s;
    }
}

// ---------------- Kernel 1: pass1 scores via WMMA f32 16x16x4 ----------------
constexpr int KC   = 128;   // K chunk staged in LDS
constexpr int NC   = D / KC;
constexpr int RPB  = 64;    // rows per block (4 waves x 16)
constexpr int PADX = 132;   // lds X row stride (floats), conflict padding
constexpr int PADR = 516;   // lds r row stride (floats), conflict padding

__global__ __launch_bounds__(128) void k_pass1(
    const float* __restrict__ X, const float* __restrict__ r,
    const float* __restrict__ cbias, const unsigned char* __restrict__ mask,
    float* __restrict__ scores)
{
    __shared__ float lds_x[2][RPB * PADX];   // 2 x 33,792 B (double buffer)
    __shared__ float lds_r[H * PADR];        // 16,512 B
    const int t    = threadIdx.x;
    const int lane = t & 31;
    const int wave = t >> 5;
    const int row0 = blockIdx.x * RPB;

    // async-stage r (H x D) into LDS once
    for (int i = t; i < H * (D / 4); i += 128) {
        const int h  = i / (D / 4);
        const int k4 = (i % (D / 4)) * 4;
        async_copy_b128(&lds_r[h * PADR + k4], r + h * D + k4);
    }
    // async-stage X chunk 0
    for (int i = t; i < RPB * (KC / 4); i += 128) {
        const int rr = i / (KC / 4);
        const int cc = (i % (KC / 4)) * 4;
        async_copy_b128(&lds_x[0][rr * PADX + cc], X + (size_t)(row0 + rr) * D + cc);
    }

    v8f acc = {};
    const int mrow  = lane & 15;          // A row within tile / B column (head id if < 8)
    const int khalf = (lane >> 4) * 2;    // K sub-offset 0 or 2
    // Columns 8..15 of the WMMA tile are never stored, so lanes with mrow>=8
    // just read a duplicate r row unconditionally -> clean ds_load_b64, no exec juggling.
    const float* rb = &lds_r[(mrow & 7) * PADR];

    for (int c = 0; c < NC; ++c) {
        const int kc = c * KC;
        wait_async0();         // chunk c (and r on first iter) has landed in LDS
        __syncthreads();       // visible block-wide; also: everyone done computing on buf (c+1)&1
        if (c + 1 < NC) {      // issue next chunk while computing this one
            const int kn = kc + KC;
            for (int i = t; i < RPB * (KC / 4); i += 128) {
                const int rr = i / (KC / 4);
                const int cc = (i % (KC / 4)) * 4;
                async_copy_b128(&lds_x[(c + 1) & 1][rr * PADX + cc],
                                X + (size_t)(row0 + rr) * D + kn + cc);
            }
        }
        const float* xa = &lds_x[c & 1][(wave * 16 + mrow) * PADX];
        #pragma unroll
        for (int kk = 0; kk < KC; kk += 4) {
            v2f a, b;
            a.x = xa[kk + khalf];
            a.y = xa[kk + khalf + 1];
            b.x = rb[kc + kk + khalf];
            b.y = rb[kc + kk + khalf + 1];
            acc = __builtin_amdgcn_wmma_f32_16x16x4_f32(
                false, a, false, b, (short)0, acc, false, false);
        }
    }

    // store: lane holds column (head) mrow, rows row0+wave*16+{0..7}(+8)
    if (mrow < H) {
        const float ch    = cbias[mrow];
        const int   rbase = row0 + wave * 16 + ((lane < 16) ? 0 : 8);
        #pragma unroll
        for (int v = 0; v < 8; ++v) {
            const int n = rbase + v;
            float s = (acc[v] + ch) * 0.125f;           // / sqrt(hd)=8
            if (!mask[n]) s = -INFINITY;
            scores[(size_t)mrow * N_TOT + n] = s;
        }
    }
}

// ---------------- Kernels 2a/2b: per-head max ----------------
__global__ __launch_bounds__(256) void k_max1(const float* __restrict__ scores,
                                              float* __restrict__ blockmax)
{
    __shared__ float sm[256];
    const int b = blockIdx.x, t = threadIdx.x;
    const int CH = N_TOT / 256;   // 512
    for (int h = 0; h < H; ++h) {
        float m = -INFINITY;
        const float* p = scores + (size_t)h * N_TOT + b * CH;
        for (int i = t; i < CH; i += 256) m = fmaxf(m, p[i]);
        sm[t] = m; __syncthreads();
        for (int s = 128; s > 0; s >>= 1) { if (t < s) sm[t] = fmaxf(sm[t], sm[t + s]); __syncthreads(); }
        if (t == 0) blockmax[h * 256 + b] = sm[0];
        __syncthreads();
    }
}

__global__ __launch_bounds__(256) void k_max2(const float* __restrict__ blockmax,
                                              float* __restrict__ gmax)
{
    __shared__ float sm[256];
    const int t = threadIdx.x;
    for (int h = 0; h < H; ++h) {
        sm[t] = blockmax[h * 256 + t]; __syncthreads();
        for (int s = 128; s > 0; s >>= 1) { if (t < s) sm[t] = fmaxf(sm[t], sm[t + s]); __syncthreads(); }
        if (t == 0) gmax[h] = sm[0];
        __syncthreads();
    }
}

// ---------------- Kernels 3a/3b: exp in place + per-head sum ----------------
__global__ __launch_bounds__(256) void k_exp1(float* __restrict__ scores,
                                              const float* __restrict__ gmax,
                                              float* __restrict__ blocksum)
{
    __shared__ float sm[256];
    const int b = blockIdx.x, t = threadIdx.x;
    const int CH = N_TOT / 256;
    for (int h = 0; h < H; ++h) {
        const float mh = gmax[h];
        float s = 0.f;
        float* p = scores + (size_t)h * N_TOT + b * CH;
        for (int i = t; i < CH; i += 256) {
            const float e = expf(p[i] - mh);   // exp(-inf)=0 handles mask
            p[i] = e;
            s += e;
        }
        sm[t] = s; __syncthreads();
        for (int k = 128; k > 0; k >>= 1) { if (t < k) sm[t] += sm[t + k]; __syncthreads(); }
        if (t == 0) blocksum[h * 256 + b] = sm[0];
        __syncthreads();
    }
}

__global__ __launch_bounds__(256) void k_sum2(const float* __restrict__ blocksum,
                                              float* __restrict__ invZ)
{
    __shared__ float sm[256];
    const int t = threadIdx.x;
    for (int h = 0; h < H; ++h) {
        sm[t] = blocksum[h * 256 + t]; __syncthreads();
        for (int s = 128; s > 0; s >>= 1) { if (t < s) sm[t] += sm[t + s]; __syncthreads(); }
        if (t == 0) invZ[h] = 1.0f / sm[0];
        __syncthreads();
    }
}

// ---------------- Kernel 4: pass2 partial t_h = sum_n e[h,n] x_n  + attention weights out ----
constexpr int P2_BLOCKS = 256;
constexpr int P2_ROWS   = N_TOT / P2_BLOCKS;   // 512

__global__ __launch_bounds__(256) void k_pass2(
    const float* __restrict__ X, const float* __restrict__ e,
    const float* __restrict__ invZ,
    float* __restrict__ partial, float* __restrict__ fw)
{
    __shared__ float els[H][256];
    const int b = blockIdx.x, t = threadIdx.x;
    const int row0 = b * P2_ROWS;

    float iz[H];
    #pragma unroll
    for (int h = 0; h < H; ++h) iz[h] = invZ[h];

    float acc0[H], acc1[H];
    #pragma unroll
    for (int h = 0; h < H; ++h) { acc0[h] = 0.f; acc1[h] = 0.f; }

    for (int chunk = 0; chunk < P2_ROWS; chunk += 256) {
        __syncthreads();
        #pragma unroll
        for (int h = 0; h < H; ++h)
            els[h][t] = e[(size_t)h * N_TOT + row0 + chunk + t];
        __syncthreads();

        for (int n = 0; n < 256; ++n) {
            const size_t base = (size_t)(row0 + chunk + n) * D;
            const float x0 = X[base + t];
            const float x1 = X[base + t + 256];
            #pragma unroll
            for (int h = 0; h < H; ++h) {
                const float w = els[h][n];
                acc0[h] = fmaf(w, x0, acc0[h]);
                acc1[h] = fmaf(w, x1, acc1[h]);
            }
        }
        // attention-weight output for row (row0+chunk+t), using staged LDS slice
        float s = 0.f;
        #pragma unroll
        for (int h = 0; h < H; ++h) s = fmaf(els[h][t], iz[h], s);
        fw[row0 + chunk + t] = s * (1.0f / (float)H);
    }

    #pragma unroll
    for (int h = 0; h < H; ++h) {
        partial[(size_t)b * (H * D) + h * D + t]       = acc0[h];
        partial[(size_t)b * (H * D) + h * D + t + 256] = acc1[h];
    }
}

// ---------------- Kernel 5: finalize -> package_embedding ----------------
__global__ __launch_bounds__(512) void k_final(
    const float* __restrict__ partial, const float* __restrict__ invZ,
    const float* __restrict__ Wv, const float* __restrict__ bv,
    const float* __restrict__ Wo, const float* __restrict__ bo,
    float* __restrict__ out)
{
    __shared__ float s_t[H * D];   // 16 KB
    __shared__ float flat[D];
    const int t = threadIdx.x;

    for (int i = t; i < H * D; i += 512) {
        float s = 0.f;
        for (int b = 0; b < P2_BLOCKS; ++b) s += partial[(size_t)b * (H * D) + i];
        s_t[i] = s;
    }
    __syncthreads();

    // weighted[h][a] = invZ[h] * (Wv[h*64+a,:] . t_h) + bv[h*64+a]
    const int h = t >> 6, a = t & 63;
    const float* wrow = Wv + (size_t)(h * HD + a) * D;
    float dot = 0.f;
    for (int j = 0; j < D; ++j) dot = fmaf(wrow[j], s_t[h * D + j], dot);
    flat[a * H + h] = dot * invZ[h] + bv[h * HD + a];   // transpose-flatten (hd,H) order
    __syncthreads();

    // package[t] = bo[t] + Wo[t,:] . flat
    const float* worow = Wo + (size_t)t * D;
    float p = 0.f;
    for (int j = 0; j < D; ++j) p = fmaf(worow[j], flat[j], p);
    out[t] = p + bo[t];
}

// ---------------- launch ----------------
extern "C" void kernel_launch(void* const* d_in, const int* in_sizes, int n_in,
                              void* d_out, int out_size, void* d_ws, size_t ws_size,
                              hipStream_t stream) {
    const float*         X    = (const float*)d_in[0];
    const unsigned char* mask = (const unsigned char*)d_in[1];
    const float*         mq   = (const float*)d_in[2];
    const float*         Wq   = (const float*)d_in[3];
    const float*         bq   = (const float*)d_in[4];
    const float*         Wk   = (const float*)d_in[5];
    const float*         bk   = (const float*)d_in[6];
    const float*         Wv   = (const float*)d_in[7];
    const float*         bv   = (const float*)d_in[8];
    const float*         Wo   = (const float*)d_in[9];
    const float*         bo   = (const float*)d_in[10];

    float* ws = (float*)d_ws;
    float* c_h      = ws + 512;            // 8
    float* gmax     = ws + 520;            // 8
    float* invZ     = ws + 528;            // 8
    float* r        = ws + 1024;           // 4096
    float* blockmax = ws + 8192;           // 2048
    float* blocksum = ws + 10240;          // 2048
    float* scores   = ws + 16384;          // 8*131072 = 1,048,576 (becomes e in place)
    float* partial  = ws + 16384 + (size_t)H * N_TOT;   // 256*4096 = 1,048,576

    float* out_pkg = (float*)d_out;        // [0..511]
    float* out_fw  = (float*)d_out + D;    // [512 .. 512+N)

    k_prep <<<1, 512, 0, stream>>>(mq, Wq, bq, Wk, bk, c_h, r);
    k_pass1<<<N_TOT / RPB, 128, 0, stream>>>(X, r, c_h, mask, scores);
    k_max1 <<<256, 256, 0, stream>>>(scores, blockmax);
    k_max2 <<<1, 256, 0, stream>>>(blockmax, gmax);
    k_exp1 <<<256, 256, 0, stream>>>(scores, gmax, blocksum);
    k_sum2 <<<1, 256, 0, stream>>>(blocksum, invZ);
    k_pass2<<<P2_BLOCKS, 256, 0, stream>>>(X, scores, invZ, partial, out_fw);
    k_final<<<1, 512, 0, stream>>>(partial, invZ, Wv, bv, Wo, bo, out_pkg);
}